// ESOLNet_14723147891347
// MI455X (gfx1250) — compile-verified
//
#include <hip/hip_runtime.h>

#define FIN 9
#define HID 64

typedef __attribute__((ext_vector_type(2))) float v2f;
typedef __attribute__((ext_vector_type(8))) float v8f;

// ---------------- utility ----------------
__global__ void zero_f32_kernel(float* __restrict__ p, int n) {
    int t = blockIdx.x * blockDim.x + threadIdx.x;
    if (t < n) p[t] = 0.0f;
}

// deg[dst] += 1 per edge
__global__ void degree_kernel(const int* __restrict__ dst, float* __restrict__ deg, int E) {
    int t = blockIdx.x * blockDim.x + threadIdx.x;
    if (t < E) atomicAdd(&deg[dst[t]], 1.0f);
}

// dinv = rsqrt(deg + 1)   (in place)
__global__ void dinv_kernel(float* __restrict__ deg, int n) {
    int t = blockIdx.x * blockDim.x + threadIdx.x;
    if (t < n) deg[t] = rsqrtf(deg[t] + 1.0f);
}

// ---------------- GEMM1: h = X @ W1   (K=9, too small for WMMA) ----------------
__global__ void gemm1_kernel(const float* __restrict__ x, const float* __restrict__ W1,
                             float* __restrict__ h, int n) {
    int t = blockIdx.x * blockDim.x + threadIdx.x;
    if (t >= n * HID) return;
    int node = t >> 6;
    int c    = t & 63;
    const float* xr = x + node * FIN;
    float s = 0.0f;
#pragma unroll
    for (int k = 0; k < FIN; ++k) s += xr[k] * W1[k * HID + c];
    h[t] = s;
}

// ---------------- GEMM2: Hout = Hin @ W  via V_WMMA_F32_16X16X4_F32 ----------------
// One wave computes a 16x16 output tile; K=64 -> 16 chained WMMAs.
// W (64x64, 16KB) staged in LDS once per block. N is a multiple of 16 (100000 = 6250*16).
__global__ void __launch_bounds__(256)
gemm_wmma_kernel(const float* __restrict__ Hin, const float* __restrict__ W,
                 float* __restrict__ Hout, int nTilesM) {
    __shared__ float sW[HID * HID];
    for (int i = threadIdx.x; i < HID * HID; i += blockDim.x) sW[i] = W[i];
    __syncthreads();

    int wave = threadIdx.x >> 5;          // 8 waves per block (wave32)
    int lane = threadIdx.x & 31;
    int tile = blockIdx.x * 8 + wave;
    int mTile = tile >> 2;                // 4 N-tiles (HID/16)
    int nTile = tile & 3;
    if (mTile >= nTilesM) return;         // uniform per wave -> EXEC stays all-1s

    int m0 = mTile * 16;
    int n0 = nTile * 16;
    int laneLo = lane & 15;
    int laneHi = lane >> 4;               // 0: K+0/1, 1: K+2/3 (A/B 32-bit fragment layout)

    const float* arow = Hin + (size_t)(m0 + laneLo) * HID;

    v8f c = {};
#pragma unroll
    for (int k0 = 0; k0 < HID; k0 += 4) {
        int kb = k0 + laneHi * 2;
        v2f a, b;
        a.x = arow[kb];
        a.y = arow[kb + 1];
        b.x = sW[kb * HID + n0 + laneLo];
        b.y = sW[(kb + 1) * HID + n0 + laneLo];
        // D = A(16x4,f32) * B(4x16,f32) + C(16x16,f32)
        c = __builtin_amdgcn_wmma_f32_16x16x4_f32(false, a, false, b,
                                                  (short)0, c, false, false);
    }

    // C/D layout: VGPR r -> row m0 + r (lanes 0-15) / m0 + 8 + r (lanes 16-31), col = n0 + laneLo
#pragma unroll
    for (int r = 0; r < 8; ++r) {
        int row = m0 + laneHi * 8 + r;
        Hout[(size_t)row * HID + n0 + laneLo] = c[r];
    }
}

// ---------------- edge scatter: agg[dst] += h[src] * dinv[src]*dinv[dst] ----------------
// One thread handles one edge x 4 channels (float4 gather, 4 f32 atomics).
__global__ void scatter_kernel(const float* __restrict__ h, const float* __restrict__ dinv,
                               const int* __restrict__ src, const int* __restrict__ dst,
                               float* __restrict__ agg, int E) {
    int t = blockIdx.x * blockDim.x + threadIdx.x;
    if (t >= E * 16) return;
    int e = t >> 4;
    int q = t & 15;
    int s = src[e];
    int d = dst[e];
    float coef = dinv[s] * dinv[d];
    float4 v = ((const float4*)(h + (size_t)s * HID))[q];
    float* ap = agg + (size_t)d * HID + q * 4;
    atomicAdd(ap + 0, v.x * coef);
    atomicAdd(ap + 1, v.y * coef);
    atomicAdd(ap + 2, v.z * coef);
    atomicAdd(ap + 3, v.w * coef);
}

// ---------------- combine: agg = relu(agg + dinv^2 * h + b)  (in place) ----------------
__global__ void combine_kernel(float* __restrict__ agg, const float* __restrict__ h,
                               const float* __restrict__ dinv, const float* __restrict__ b,
                               int n) {
    int t = blockIdx.x * blockDim.x + threadIdx.x;
    if (t >= n * HID) return;
    int node = t >> 6;
    int c    = t & 63;
    float di = dinv[node];
    float v = agg[t] + di * di * h[t] + b[c];
    agg[t] = v > 0.0f ? v : 0.0f;
}

// ---------------- global max pool (post-ReLU values >= 0 -> uint-bit atomicMax valid) ----
__global__ void pool_kernel(const float* __restrict__ h, const int* __restrict__ batch,
                            unsigned int* __restrict__ pool, int n) {
    int t = blockIdx.x * blockDim.x + threadIdx.x;
    if (t >= n * 16) return;
    int node = t >> 4;
    int q    = t & 15;
    int g = batch[node];
    float4 v = ((const float4*)(h + (size_t)node * HID))[q];
    unsigned int* p = pool + (size_t)g * HID + q * 4;
    atomicMax(p + 0, __float_as_uint(v.x));
    atomicMax(p + 1, __float_as_uint(v.y));
    atomicMax(p + 2, __float_as_uint(v.z));
    atomicMax(p + 3, __float_as_uint(v.w));
}

// ---------------- out = pooled @ Wout + bout  (one block of 64 threads per graph) -------
__global__ void outproj_kernel(const float* __restrict__ pool, const float* __restrict__ Wout,
                               const float* __restrict__ bout, float* __restrict__ out) {
    __shared__ float red[HID];
    int g = blockIdx.x;
    int t = threadIdx.x;   // 64 threads
    red[t] = pool[(size_t)g * HID + t] * Wout[t];
    __syncthreads();
#pragma unroll
    for (int s = 32; s > 0; s >>= 1) {
        if (t < s) red[t] += red[t + s];
        __syncthreads();
    }
    if (t == 0) out[g] = red[0] + bout[0];
}

// ---------------- launcher ----------------
extern "C" void kernel_launch(void* const* d_in, const int* in_sizes, int n_in,
                              void* d_out, int out_size, void* d_ws, size_t ws_size,
                              hipStream_t stream) {
    const float* x     = (const float*)d_in[0];
    const int*   ei    = (const int*)d_in[1];   // [2, E]
    const int*   batch = (const int*)d_in[2];
    const float* W1    = (const float*)d_in[4];
    const float* b1    = (const float*)d_in[5];
    const float* W2    = (const float*)d_in[6];
    const float* b2    = (const float*)d_in[7];
    const float* Wout  = (const float*)d_in[8];
    const float* bout  = (const float*)d_in[9];
    float*       out   = (float*)d_out;

    const int N = in_sizes[0] / FIN;   // 100000
    const int E = in_sizes[1] / 2;     // 3200000
    const int G = out_size;            // 1024

    const int* src = ei;
    const int* dst = ei + E;

    // workspace layout (floats): [dinv(N)] [hbuf(N*HID)] [agg(N*HID)] [pool(G*HID)]
    float* ws   = (float*)d_ws;
    size_t nPad = ((size_t)N + 255) & ~(size_t)255;
    float* dinv = ws;
    float* hbuf = ws + nPad;
    float* agg  = hbuf + (size_t)N * HID;
    float* pool = agg  + (size_t)N * HID;

    const int TB = 256;
    const int nNH   = N * HID;
    const int nTilesM = N / 16;                       // 6250
    const int gemmBlocks = (nTilesM * 4 + 7) / 8;     // 8 waves/block, 4 N-tiles

    // degree -> dinv
    zero_f32_kernel<<<(N + TB - 1) / TB, TB, 0, stream>>>(dinv, N);
    degree_kernel<<<(E + TB - 1) / TB, TB, 0, stream>>>(dst, dinv, E);
    dinv_kernel<<<(N + TB - 1) / TB, TB, 0, stream>>>(dinv, N);

    // ---- layer 1 ----
    gemm1_kernel<<<(nNH + TB - 1) / TB, TB, 0, stream>>>(x, W1, hbuf, N);
    zero_f32_kernel<<<(nNH + TB - 1) / TB, TB, 0, stream>>>(agg, nNH);
    scatter_kernel<<<(E * 16 + TB - 1) / TB, TB, 0, stream>>>(hbuf, dinv, src, dst, agg, E);
    combine_kernel<<<(nNH + TB - 1) / TB, TB, 0, stream>>>(agg, hbuf, dinv, b1, N);
    // agg now holds h1 (post-ReLU)

    // ---- layer 2 ----
    gemm_wmma_kernel<<<gemmBlocks, TB, 0, stream>>>(agg, W2, hbuf, nTilesM);
    zero_f32_kernel<<<(nNH + TB - 1) / TB, TB, 0, stream>>>(agg, nNH);
    scatter_kernel<<<(E * 16 + TB - 1) / TB, TB, 0, stream>>>(hbuf, dinv, src, dst, agg, E);
    combine_kernel<<<(nNH + TB - 1) / TB, TB, 0, stream>>>(agg, hbuf, dinv, b2, N);
    // agg now holds h2 (post-ReLU, all >= 0)

    // ---- pooling + output projection ----
    zero_f32_kernel<<<(G * HID + TB - 1) / TB, TB, 0, stream>>>(pool, G * HID);
    pool_kernel<<<(N * 16 + TB - 1) / TB, TB, 0, stream>>>(agg, batch, (unsigned int*)pool, N);
    outproj_kernel<<<G, HID, 0, stream>>>(pool, Wout, bout, out);
}